// Morton_decode_69312182223578
// MI455X (gfx1250) — compile-verified
//
#include <hip/hip_runtime.h>
#include <stdint.h>

// ---------------------------------------------------------------------------
// Morton (Z-order) decode: out[b][c][i][j] = x[b][c][ morton(i,j) ]
// i = odd bits of the morton index, j = even bits.
// Strategy: 64x64 output tile == contiguous 4096-float input chunk.
//   Phase 1: async global->LDS copy of the contiguous chunk
//            (GLOBAL_LOAD_ASYNC_TO_LDS_B128, tracked by ASYNCcnt).
//   Phase 2: morton-decoded LDS reads (swizzled to avoid bank conflicts),
//            fully-coalesced float4 stores to the output tile.
// ---------------------------------------------------------------------------

#if defined(__has_builtin)
# if __has_builtin(__builtin_amdgcn_global_load_async_to_lds_b128)
#  define MD_HAVE_ASYNC_LDS 1
# endif
# if __has_builtin(__builtin_amdgcn_s_wait_asynccnt)
#  define MD_HAVE_WAIT_ASYNC 1
# endif
#endif

typedef int v4i __attribute__((ext_vector_type(4)));

// Spread low 8 bits of v so bit b lands at bit 2b (interleave helper).
__device__ __forceinline__ uint32_t md_spread8(uint32_t v) {
  v &= 0xFFu;
  v = (v | (v << 4)) & 0x0F0Fu;
  v = (v | (v << 2)) & 0x3333u;
  v = (v | (v << 1)) & 0x5555u;
  return v;
}

// LDS swizzle: insert 8 pad dwords per 64-dword block so morton bits 6/8
// contribute to the bank index (64 banks x 4B). Accesses used below never
// straddle a pad boundary (16B chunks start at m%64 in {0..60} multiples of 4;
// 8B chunks start at even m, and m+4 is re-swizzled separately).
__device__ __forceinline__ uint32_t md_swz(uint32_t m) {
  return m + ((m >> 6) << 3);
}

__global__ __launch_bounds__(256)
void morton_decode_kernel(const float* __restrict__ x, float* __restrict__ out) {
  __shared__ float lds[4600];  // 4096 data dwords + 63*8 pad dwords

  const uint32_t tid   = threadIdx.x;      // 0..255
  const uint32_t bid   = blockIdx.x;       // plane*16 + tile
  const uint32_t plane = bid >> 4;         // b*64 + c  (512 planes)
  const uint32_t tile  = bid & 15u;        // 16 tiles of 64x64 per plane

  const uint64_t planeBase = (uint64_t)plane * 65536u;
  const float* __restrict__ src = x + planeBase + (uint64_t)tile * 4096u;

  // ---- Phase 1: contiguous morton chunk -> LDS (16 floats per thread) ----
#if defined(MD_HAVE_ASYNC_LDS)
  #pragma unroll
  for (int i = 0; i < 4; ++i) {
    const uint32_t m = (uint32_t)i * 1024u + tid * 4u;   // linear morton dword
    const uint32_t a = md_swz(m);
    __builtin_amdgcn_global_load_async_to_lds_b128(
        (v4i*)(src + m),
        (v4i*)(&lds[a]),
        0, 0);
  }
# if defined(MD_HAVE_WAIT_ASYNC)
  __builtin_amdgcn_s_wait_asynccnt(0);
# else
  asm volatile("s_wait_asynccnt 0" ::: "memory");
# endif
#else
  #pragma unroll
  for (int i = 0; i < 4; ++i) {
    const uint32_t m = (uint32_t)i * 1024u + tid * 4u;
    const uint32_t a = md_swz(m);
    const float4 v = *(const float4*)(src + m);
    *(float4*)(&lds[a]) = v;
  }
#endif
  __syncthreads();

  // ---- Phase 2: morton-decode from LDS, coalesced float4 stores ----
  // tile index bits: morton bits 12..15 of the full 16-bit index.
  const uint32_t ti = ((tile >> 1) & 1u) | (((tile >> 3) & 1u) << 1);  // odd bits
  const uint32_t tj = (tile & 1u) | (((tile >> 2) & 1u) << 1);         // even bits

  float* __restrict__ dst = out + planeBase;

  const uint32_t jj = (tid & 15u) * 4u;   // local col, multiple of 4
  const uint32_t sj = md_spread8(jj);     // even-bit spread of local col

  #pragma unroll
  for (int k = 0; k < 4; ++k) {
    const uint32_t r  = (uint32_t)k * 16u + (tid >> 4);  // local row 0..63
    const uint32_t m  = sj | (md_spread8(r) << 1);       // source morton index
    // (r, jj..jj+3) live at lds morton indices m, m+1, m+4, m+5
    const uint32_t a0 = md_swz(m);
    const uint32_t a1 = md_swz(m + 4u);
    const float2 lo = *(const float2*)(&lds[a0]);
    const float2 hi = *(const float2*)(&lds[a1]);
    const float4 v = make_float4(lo.x, lo.y, hi.x, hi.y);

    const uint32_t row = ti * 64u + r;
    const uint32_t col = tj * 64u + jj;
    // 16 consecutive lanes -> 256B contiguous row segment: fully coalesced.
    *(float4*)(dst + (uint64_t)row * 256u + col) = v;
  }
}

extern "C" void kernel_launch(void* const* d_in, const int* in_sizes, int n_in,
                              void* d_out, int out_size, void* d_ws, size_t ws_size,
                              hipStream_t stream) {
  (void)in_sizes; (void)n_in; (void)out_size; (void)d_ws; (void)ws_size;
  const float* x = (const float*)d_in[0];
  float* out = (float*)d_out;
  // 512 planes * 16 tiles = 8192 blocks, 256 threads (8 waves of 32).
  morton_decode_kernel<<<8192, 256, 0, stream>>>(x, out);
}